// OFDM_5282809774295
// MI455X (gfx1250) — compile-verified
//
#include <hip/hip_runtime.h>
#include <math.h>

// ---------------- problem constants ----------------
#define NB   64
#define PP   2
#define SS   64
#define MM   4096
#define KK   64
#define LL   16
#define NPIL 2
#define SEG    (SS + NPIL)        // 66 segments per (n,p) stream
#define SEGLEN (MM + KK)          // 4160
#define ROWLEN (SEG * SEGLEN)     // 274560 samples per (n,p) stream
#define NROWS  (NB * PP)          // 128 streams
#define TWO_PI 6.283185307179586f
#define CLIP_RATIO 1.2f

typedef float v2f __attribute__((ext_vector_type(2)));
typedef float v8f __attribute__((ext_vector_type(8)));
typedef int   v4i_gcc __attribute__((vector_size(16)));
typedef __attribute__((address_space(1))) v4i_gcc* gptr_v4i;
typedef __attribute__((address_space(3))) v4i_gcc* lptr_v4i;

// ---------------- workspace layout (float offsets) ----------------
static constexpr size_t STREAMSZ = (size_t)NROWS * ROWLEN;   // 35,143,680
static constexpr size_t WS_XCRE  = 0;
static constexpr size_t WS_XCIM  = STREAMSZ;
static constexpr size_t WS_YRE   = 2 * STREAMSZ;
static constexpr size_t WS_YIM   = 3 * STREAMSZ;
static constexpr size_t WS_STATS = 4 * STREAMSZ;
// stats per row (8 f32): 0 sum|x|^2, 1 max|x|^2, 2 sum|xc|^2, 3 max|xc|^2,
//                        4 sum|y|^2, 5 clip threshold

// ---------------- output layout (float offsets, complex interleaved) ------
static constexpr size_t OUT_IP     = 0;                                     // info_pilot (64,2,2,4096) c64
static constexpr size_t OUT_IS     = OUT_IP + (size_t)NB*PP*NPIL*MM*2;      // info_sig   (64,2,64,4096) c64
static constexpr size_t OUT_HT     = OUT_IS + (size_t)NB*PP*SS*MM*2;        // H_t        (64,2,16) c64
static constexpr size_t OUT_NPWR   = OUT_HT + (size_t)NB*PP*LL*2;           // noise_pwr  (64,2,1)
static constexpr size_t OUT_PAPR   = OUT_NPWR + NROWS;                      // papr       (64,2)
static constexpr size_t OUT_PAPRCP = OUT_PAPR + NROWS;                      // papr_cp    (64,2)

// ---------------- async global->LDS staging (CDNA5 path) ------------------
__device__ __forceinline__ void stage_row_to_lds(const float* __restrict__ gsrc,
                                                 float* lds_dst, int tid)
{
#if __has_builtin(__builtin_amdgcn_global_load_async_to_lds_b128)
    // GLOBAL_LOAD_ASYNC_TO_LDS_B128: 16B per lane, tracked with ASYNCcnt.
    for (int j = tid * 4; j < MM; j += 512 * 4) {
        __builtin_amdgcn_global_load_async_to_lds_b128(
            (gptr_v4i)(gsrc + j),
            (lptr_v4i)(lds_dst + j),
            0, 0);
    }
#else
    for (int j = tid; j < MM; j += 512) lds_dst[j] = gsrc[j];
#endif
}

__device__ __forceinline__ void async_stage_wait()
{
#if __has_builtin(__builtin_amdgcn_global_load_async_to_lds_b128)
  #if __has_builtin(__builtin_amdgcn_s_wait_asynccnt)
    __builtin_amdgcn_s_wait_asynccnt(0);
  #else
    asm volatile("s_wait_asynccnt 0x0" ::: "memory");
  #endif
#endif
}

// =====================================================================
// 4096-point (I)FFT via radix-64x64 tensor-core decomposition.
//   x[64*n1+n2]  viewed as 64x64 matrix A[n1][n2] (row-major in LDS)
//   stage 1: Y = F64 * A   (F64[a][b] = exp(sign*2pi*i*a*b/64), LDS table)
//   twiddle: Z[k1][n2] = Y[k1][n2] * exp(sign*2pi*i*k1*n2/4096)
//   stage 2: O = Z * F64  ;  X[k1 + 64*k2] = scale * O[k1][k2]
// One workgroup = 512 threads = 16 wave32; wave w owns 16x16 tile (w/4,w%4).
// Complex matmul = 4 real v_wmma_f32_16x16x4_f32 accumulations per k-step.
// F64 fragments come from a precomputed LDS table (ds_load) instead of
// per-kstep sincos: 8x fewer transcendentals in the hot loop.
// =====================================================================
__device__ __forceinline__ void fft4096_core(float* xr, float* xi,
                                             float* zr, float* zi,
                                             const float* Fr, const float* Fi,
                                             float sign, float scale)
{
    const int tid   = threadIdx.x;
    const int lane  = tid & 31;
    const int wave  = tid >> 5;            // 0..15
    const int tm    = wave >> 2;
    const int tn    = wave & 3;
    const int mrow  = tm * 16 + (lane & 15);   // A-fragment row (M)
    const int ncol  = tn * 16 + (lane & 15);   // B/C-fragment col (N)
    const int khalf = (lane >> 4) << 1;        // K sub-offset per A/B layout
    const float wN  = sign * (TWO_PI / 4096.0f);

    // ---------------- stage 1: A = F64 table, B = x ----------------
    {
        v8f rr = {}, ii = {}, ri = {}, ir = {};
        for (int k0 = 0; k0 < 64; k0 += 4) {
            const int ka = k0 + khalf;
            v2f Ar, Ai, Br, Bi;
            Ar.x = Fr[mrow * 64 + ka]; Ar.y = Fr[mrow * 64 + ka + 1];
            Ai.x = Fi[mrow * 64 + ka]; Ai.y = Fi[mrow * 64 + ka + 1];
            Br.x = xr[ka * 64 + ncol]; Br.y = xr[(ka + 1) * 64 + ncol];
            Bi.x = xi[ka * 64 + ncol]; Bi.y = xi[(ka + 1) * 64 + ncol];
            rr = __builtin_amdgcn_wmma_f32_16x16x4_f32(false, Ar, false, Br, (short)0, rr, false, false);
            ii = __builtin_amdgcn_wmma_f32_16x16x4_f32(false, Ai, false, Bi, (short)0, ii, false, false);
            ri = __builtin_amdgcn_wmma_f32_16x16x4_f32(false, Ar, false, Bi, (short)0, ri, false, false);
            ir = __builtin_amdgcn_wmma_f32_16x16x4_f32(false, Ai, false, Br, (short)0, ir, false, false);
        }
#pragma unroll
        for (int r = 0; r < 8; ++r) {
            const int k1 = tm * 16 + r + ((lane >> 4) << 3);      // C layout: M = r (+8 hi half)
            const float re = rr[r] - ii[r];
            const float im = ri[r] + ir[r];
            float ts, tc;
            __sincosf(wN * (float)((k1 * ncol) & 4095), &ts, &tc);
            zr[k1 * 64 + ncol] = re * tc - im * ts;
            zi[k1 * 64 + ncol] = re * ts + im * tc;
        }
    }
    __syncthreads();
    // ---------------- stage 2: A = Z, B = F64 table ----------------
    {
        v8f rr = {}, ii = {}, ri = {}, ir = {};
        for (int k0 = 0; k0 < 64; k0 += 4) {
            const int ka = k0 + khalf;
            v2f Ar, Ai, Br, Bi;
            Ar.x = zr[mrow * 64 + ka]; Ar.y = zr[mrow * 64 + ka + 1];
            Ai.x = zi[mrow * 64 + ka]; Ai.y = zi[mrow * 64 + ka + 1];
            Br.x = Fr[ka * 64 + ncol]; Br.y = Fr[(ka + 1) * 64 + ncol];   // F symmetric
            Bi.x = Fi[ka * 64 + ncol]; Bi.y = Fi[(ka + 1) * 64 + ncol];
            rr = __builtin_amdgcn_wmma_f32_16x16x4_f32(false, Ar, false, Br, (short)0, rr, false, false);
            ii = __builtin_amdgcn_wmma_f32_16x16x4_f32(false, Ai, false, Bi, (short)0, ii, false, false);
            ri = __builtin_amdgcn_wmma_f32_16x16x4_f32(false, Ar, false, Bi, (short)0, ri, false, false);
            ir = __builtin_amdgcn_wmma_f32_16x16x4_f32(false, Ai, false, Br, (short)0, ir, false, false);
        }
#pragma unroll
        for (int r = 0; r < 8; ++r) {
            const int k1 = tm * 16 + r + ((lane >> 4) << 3);
            xr[k1 + 64 * ncol] = scale * (rr[r] - ii[r]);
            xi[k1 + 64 * ncol] = scale * (ri[r] + ir[r]);
        }
    }
}

// Build the DFT-64 table (overlaps with the async input DMA).
__device__ __forceinline__ void build_f64_table(float* Fr, float* Fi, float sign, int tid)
{
    const float w64 = sign * (TWO_PI / 64.0f);
    for (int j = tid; j < 4096; j += 512) {
        const int a = j >> 6, b = j & 63;
        float s, c;
        __sincosf(w64 * (float)((a * b) & 63), &s, &c);
        Fr[j] = c; Fi[j] = s;
    }
}

// ============ IFFT of x rows, write into stream with cyclic prefix ========
__global__ void __launch_bounds__(512)
ifft_kernel(const float* __restrict__ xre, const float* __restrict__ xim,
            float* __restrict__ xcre, float* __restrict__ xcim)
{
    extern __shared__ float smem[];
    float* xr = smem;
    float* xi = smem + 4096;
    float* zr = smem + 8192;
    float* zi = smem + 12288;
    float* Fr = smem + 16384;
    float* Fi = smem + 20480;
    const int row = blockIdx.x;              // (n*P+p)*S + s
    const int tid = threadIdx.x;
    const size_t src = (size_t)row * MM;

    stage_row_to_lds(xre + src, xr, tid);    // async DMA global -> LDS
    stage_row_to_lds(xim + src, xi, tid);
    build_f64_table(Fr, Fi, 1.0f, tid);      // VALU work overlaps the DMA
    async_stage_wait();
    __syncthreads();

    fft4096_core(xr, xi, zr, zi, Fr, Fi, 1.0f, 1.0f / 4096.0f);
    __syncthreads();

    const int np = row >> 6;
    const int s  = row & 63;
    const size_t base = ((size_t)np * SEG + NPIL + s) * SEGLEN;
    for (int j = tid; j < MM; j += 512) {
        xcre[base + KK + j] = xr[j];
        xcim[base + KK + j] = xi[j];
    }
    if (tid < KK) {                          // cyclic prefix
        xcre[base + tid] = xr[MM - KK + tid];
        xcim[base + tid] = xi[MM - KK + tid];
    }
}

// ============ broadcast pilot_cp into segments 0..NPIL-1 of each stream ===
__global__ void pilot_kernel(const float* __restrict__ pr, const float* __restrict__ pi,
                             float* __restrict__ xcre, float* __restrict__ xcim)
{
    const size_t total = (size_t)NB * PP * NPIL * SEGLEN;
    size_t i = (size_t)blockIdx.x * blockDim.x + threadIdx.x;
    if (i >= total) return;
    const int t = (int)(i % SEGLEN);
    size_t q = i / SEGLEN;
    const int j0 = (int)(q % NPIL); q /= NPIL;
    const int p  = (int)(q % PP);
    const int n  = (int)(q / PP);
    const size_t dst = (((size_t)(n * PP + p)) * SEG + j0) * SEGLEN + t;
    const size_t sp  = ((size_t)p * NPIL + j0) * SEGLEN + t;
    xcre[dst] = pr[sp];
    xcim[dst] = pi[sp];
}

// ============ per-row sum/max of |x|^2 ====================================
__global__ void reduce1_kernel(const float* __restrict__ xcre, const float* __restrict__ xcim,
                               float* __restrict__ stats)
{
    const int row = blockIdx.y;
    const size_t base = (size_t)row * ROWLEN;
    float lsum = 0.f, lmax = 0.f;
    for (int i = blockIdx.x * blockDim.x + threadIdx.x; i < ROWLEN; i += gridDim.x * blockDim.x) {
        const float a = xcre[base + i], b = xcim[base + i];
        const float p = a * a + b * b;
        lsum += p; lmax = fmaxf(lmax, p);
    }
    __shared__ float ssum[256], smax[256];
    ssum[threadIdx.x] = lsum; smax[threadIdx.x] = lmax;
    __syncthreads();
    for (int o = 128; o > 0; o >>= 1) {
        if (threadIdx.x < o) {
            ssum[threadIdx.x] += ssum[threadIdx.x + o];
            smax[threadIdx.x] = fmaxf(smax[threadIdx.x], smax[threadIdx.x + o]);
        }
        __syncthreads();
    }
    if (threadIdx.x == 0) {
        atomicAdd(&stats[row * 8 + 0], ssum[0]);
        atomicMax((unsigned int*)&stats[row * 8 + 1], __float_as_uint(smax[0]));
    }
}

// ============ papr + clip threshold =======================================
__global__ void stats1_kernel(float* __restrict__ stats, float* __restrict__ out)
{
    const int row = threadIdx.x;
    if (row >= NROWS) return;
    const float mean = stats[row * 8 + 0] / (float)ROWLEN;
    out[OUT_PAPR + row] = 10.0f * log10f(stats[row * 8 + 1] / mean);
    stats[row * 8 + 5] = sqrtf(mean) * CLIP_RATIO;
}

// ============ in-place clipping + clipped stats ===========================
__global__ void clip_kernel(float* __restrict__ xcre, float* __restrict__ xcim,
                            float* __restrict__ stats)
{
    const int row = blockIdx.y;
    const size_t base = (size_t)row * ROWLEN;
    const float thr = stats[row * 8 + 5];
    float lsum = 0.f, lmax = 0.f;
    for (int i = blockIdx.x * blockDim.x + threadIdx.x; i < ROWLEN; i += gridDim.x * blockDim.x) {
        float a = xcre[base + i], b = xcim[base + i];
        const float amp = sqrtf(a * a + b * b);
        const float sc = fminf(thr / amp, 1.0f);     // amp==0 -> inf/NaN -> 1
        a *= sc; b *= sc;
        xcre[base + i] = a; xcim[base + i] = b;
        const float p = a * a + b * b;
        lsum += p; lmax = fmaxf(lmax, p);
    }
    __shared__ float ssum[256], smax[256];
    ssum[threadIdx.x] = lsum; smax[threadIdx.x] = lmax;
    __syncthreads();
    for (int o = 128; o > 0; o >>= 1) {
        if (threadIdx.x < o) {
            ssum[threadIdx.x] += ssum[threadIdx.x + o];
            smax[threadIdx.x] = fmaxf(smax[threadIdx.x], smax[threadIdx.x + o]);
        }
        __syncthreads();
    }
    if (threadIdx.x == 0) {
        atomicAdd(&stats[row * 8 + 2], ssum[0]);
        atomicMax((unsigned int*)&stats[row * 8 + 3], __float_as_uint(smax[0]));
    }
}

__global__ void stats2_kernel(const float* __restrict__ stats, float* __restrict__ out)
{
    const int row = threadIdx.x;
    if (row >= NROWS) return;
    const float mean = stats[row * 8 + 2] / (float)ROWLEN;
    out[OUT_PAPRCP + row] = 10.0f * log10f(stats[row * 8 + 3] / mean);
}

// ============ 16-tap complex FIR with LDS halo + |y|^2 reduction ==========
__global__ void fir_kernel(const float* __restrict__ xcre, const float* __restrict__ xcim,
                           const float* __restrict__ cofr, const float* __restrict__ cofi,
                           float* __restrict__ yre, float* __restrict__ yim,
                           float* __restrict__ stats)
{
    const int row = blockIdx.y;
    const int start = blockIdx.x * 256;
    const size_t base = (size_t)row * ROWLEN;
    __shared__ float sr[256 + LL - 1], si[256 + LL - 1];
    __shared__ float tr[LL], ti[LL];
    __shared__ float red[256];
    const int tid = threadIdx.x;
    if (tid < LL) { tr[tid] = cofr[row * LL + tid]; ti[tid] = cofi[row * LL + tid]; }
    for (int t = tid; t < 256 + LL - 1; t += 256) {
        const int pos = start - (LL - 1) + t;
        float a = 0.f, b = 0.f;
        if (pos >= 0 && pos < ROWLEN) {
            __builtin_prefetch(&xcre[base + pos], 0, 1);
            a = xcre[base + pos]; b = xcim[base + pos];
        }
        sr[t] = a; si[t] = b;
    }
    __syncthreads();
    const int n = start + tid;
    float p = 0.f;
    if (n < ROWLEN) {
        float yr = 0.f, yi = 0.f;
#pragma unroll
        for (int k = 0; k < LL; ++k) {
            const float xr_ = sr[tid + LL - 1 - k];
            const float xi_ = si[tid + LL - 1 - k];
            yr += tr[k] * xr_ - ti[k] * xi_;
            yi += tr[k] * xi_ + ti[k] * xr_;
        }
        yre[base + n] = yr; yim[base + n] = yi;
        p = yr * yr + yi * yi;
    }
    red[tid] = p; __syncthreads();
    for (int o = 128; o > 0; o >>= 1) { if (tid < o) red[tid] += red[tid + o]; __syncthreads(); }
    if (tid == 0) atomicAdd(&stats[row * 8 + 4], red[0]);
}

__global__ void stats3_kernel(const float* __restrict__ stats, const int* __restrict__ snr,
                              float* __restrict__ out)
{
    const int row = threadIdx.x;
    if (row >= NROWS) return;
    const float pwr = stats[row * 8 + 4] / (float)ROWLEN;
    out[OUT_NPWR + row] = pwr * exp10f(-(float)snr[0] / 10.0f);
}

// ============ H_t : 16-point DFT of taps ==================================
__global__ void ht_kernel(const float* __restrict__ cofr, const float* __restrict__ cofi,
                          float* __restrict__ out)
{
    const int g = blockIdx.x * blockDim.x + threadIdx.x;
    if (g >= NROWS * LL) return;
    const int row = g >> 4, k = g & 15;
    float accr = 0.f, acci = 0.f;
    for (int j = 0; j < LL; ++j) {
        float s, c;
        __sincosf(-(TWO_PI / 16.0f) * (float)((j * k) & 15), &s, &c);
        const float a = cofr[row * LL + j], b = cofi[row * LL + j];
        accr += a * c - b * s;
        acci += a * s + b * c;
    }
    out[OUT_HT + 2 * (size_t)g]     = accr;
    out[OUT_HT + 2 * (size_t)g + 1] = acci;
}

// ============ forward FFT of y segments -> d_out (interleaved complex) ====
__global__ void __launch_bounds__(512)
fft_kernel(const float* __restrict__ yre, const float* __restrict__ yim,
           float* __restrict__ out)
{
    extern __shared__ float smem[];
    float* xr = smem;
    float* xi = smem + 4096;
    float* zr = smem + 8192;
    float* zi = smem + 12288;
    float* Fr = smem + 16384;
    float* Fi = smem + 20480;
    const int b   = blockIdx.x;          // 0..NROWS*SEG-1
    const int np  = b / SEG;
    const int seg = b % SEG;
    const int tid = threadIdx.x;
    const size_t base = ((size_t)np * SEG + seg) * SEGLEN + KK;   // drop CP

    stage_row_to_lds(yre + base, xr, tid);   // async DMA global -> LDS
    stage_row_to_lds(yim + base, xi, tid);
    build_f64_table(Fr, Fi, -1.0f, tid);     // overlaps the DMA
    async_stage_wait();
    __syncthreads();

    fft4096_core(xr, xi, zr, zi, Fr, Fi, -1.0f, 1.0f);
    __syncthreads();

    size_t obase;
    if (seg < NPIL) obase = OUT_IP + (((size_t)np * NPIL + seg) * MM) * 2;
    else            obase = OUT_IS + (((size_t)np * SS + (seg - NPIL)) * MM) * 2;
    for (int j = tid; j < MM; j += 512) {
        out[obase + 2 * j]     = xr[j];
        out[obase + 2 * j + 1] = xi[j];
    }
}

__global__ void init_stats_kernel(float* __restrict__ stats)
{
    const int i = blockIdx.x * blockDim.x + threadIdx.x;
    if (i < NROWS * 8) stats[i] = 0.f;
}

// =====================================================================
extern "C" void kernel_launch(void* const* d_in, const int* in_sizes, int n_in,
                              void* d_out, int out_size, void* d_ws, size_t ws_size,
                              hipStream_t stream)
{
    (void)in_sizes; (void)n_in; (void)out_size; (void)ws_size;
    const float* xre  = (const float*)d_in[0];
    const float* xim  = (const float*)d_in[1];
    const float* cofr = (const float*)d_in[2];
    const float* cofi = (const float*)d_in[3];
    const float* pr   = (const float*)d_in[4];
    const float* pi   = (const float*)d_in[5];
    const int*   snr  = (const int*)d_in[6];
    float* out = (float*)d_out;
    float* ws  = (float*)d_ws;

    float* xcre  = ws + WS_XCRE;
    float* xcim  = ws + WS_XCIM;
    float* yre   = ws + WS_YRE;
    float* yim   = ws + WS_YIM;
    float* stats = ws + WS_STATS;

    const size_t ldsBytes = 6 * 4096 * sizeof(float);   // x, z, F64 table: 96 KB

    init_stats_kernel<<<4, 256, 0, stream>>>(stats);

    const size_t pilotTotal = (size_t)NB * PP * NPIL * SEGLEN;
    pilot_kernel<<<(unsigned)((pilotTotal + 255) / 256), 256, 0, stream>>>(pr, pi, xcre, xcim);

    ifft_kernel<<<NB * PP * SS, 512, ldsBytes, stream>>>(xre, xim, xcre, xcim);

    reduce1_kernel<<<dim3(135, NROWS), 256, 0, stream>>>(xcre, xcim, stats);
    stats1_kernel<<<1, 128, 0, stream>>>(stats, out);
    clip_kernel<<<dim3(135, NROWS), 256, 0, stream>>>(xcre, xcim, stats);
    stats2_kernel<<<1, 128, 0, stream>>>(stats, out);

    fir_kernel<<<dim3((ROWLEN + 255) / 256, NROWS), 256, 0, stream>>>(
        xcre, xcim, cofr, cofi, yre, yim, stats);
    stats3_kernel<<<1, 128, 0, stream>>>(stats, snr, out);

    ht_kernel<<<(NROWS * LL + 255) / 256, 256, 0, stream>>>(cofr, cofi, out);

    fft_kernel<<<NROWS * SEG, 512, ldsBytes, stream>>>(yre, yim, out);
}